// GaussianMixtureLoss_3779571220769
// MI455X (gfx1250) — compile-verified
//
#include <hip/hip_runtime.h>
#include <math.h>

typedef __attribute__((ext_vector_type(2))) float v2f;
typedef __attribute__((ext_vector_type(8))) float v8f;

#define BATCH  4
#define NPTS   4096
#define NMIX   4096
#define WAVES_PER_BLOCK 8
#define BLOCK_THREADS   256
#define PTS_PER_BLOCK   (WAVES_PER_BLOCK * 16)       // 128 points per block
#define BLOCKS_PER_BATCH (NPTS / PTS_PER_BLOCK)      // 32
#define NTILES (NMIX / 16)                           // 256 column tiles

__global__ void gmm_zero_out(float* out) {
    if (threadIdx.x == 0) out[0] = 0.0f;
}

__global__ __launch_bounds__(BLOCK_THREADS)
void gmm_loss_kernel(const float* __restrict__ pred,
                     const float* __restrict__ gt,
                     float* __restrict__ out)
{
    // Augmented means for this batch: (x, y, z, -|g|^2/2) -> 64 KB LDS
    __shared__ float4 smem[NMIX];

    const int b      = blockIdx.x / BLOCKS_PER_BATCH;
    const int blkInB = blockIdx.x % BLOCKS_PER_BATCH;
    const int tid    = threadIdx.x;

    const float* gtb = gt + (size_t)b * NMIX * 3;
    for (int m = tid; m < NMIX; m += BLOCK_THREADS) {
        float gx = gtb[3 * m + 0];
        float gy = gtb[3 * m + 1];
        float gz = gtb[3 * m + 2];
        float4 v;
        v.x = gx; v.y = gy; v.z = gz;
        v.w = -0.5f * (gx * gx + gy * gy + gz * gz);
        smem[m] = v;
    }
    __syncthreads();

    const int wave = tid >> 5;
    const int lane = tid & 31;
    const int half = lane >> 4;     // 0: K=0,1 half   1: K=2,3 half
    const int lid  = lane & 15;

    const int tileBase = blkInB * PTS_PER_BLOCK + wave * 16;
    const int n        = tileBase + lid;

    // Each lane loads full xyz of point (lane & 15); both halves see same point.
    const float* pp = pred + ((size_t)b * NPTS + n) * 3;
    const float px = pp[0], py = pp[1], pz = pp[2];

    const float L2E    = 1.4426950408889634f;   // log2(e)
    const float CONSTV = 0.9189385332046727f;   // 0.5*log(2*pi), sigma=1

    // Fold the exp->exp2 conversion into the WMMA: scale A and C by log2(e),
    // so D = log2(e) * log_prob and v_exp_f32 (exp2) needs no pre-multiply.
    // A fragment (16x4 f32): lanes 0-15 -> (K0,K1)=L2E*(x,y); lanes 16-31 -> (K2,K3)=(L2E*z, L2E)
    v2f a;
    a.x = L2E * (half ? pz : px);
    a.y = half ? L2E : L2E * py;

    const float rowbias = L2E * (-CONSTV - 0.5f * (px * px + py * py + pz * pz));

    // C fragment: VGPR i holds row (i + 8*half); broadcast that row's bias.
    v8f c0;
#pragma unroll
    for (int i = 0; i < 8; ++i)
        c0[i] = __shfl(rowbias, i + (half << 3), 32);

    float acc[8];
#pragma unroll
    for (int i = 0; i < 8; ++i) acc[i] = 0.0f;

    // B fragment per lane: lanes 0-15 read (x,y), lanes 16-31 read (z, -|g|^2/2)
    const char* sbase = (const char*)smem;
    const int   bofs  = (half << 3);            // byte offset 0 or 8

#pragma unroll 2
    for (int t = 0; t < NTILES; ++t) {
        v2f bv = *(const v2f*)(sbase + (((t << 4) + lid) << 4) + bofs);

        // D = log2(e) * (-CONST - ||p-g||^2 / 2)  (scaled log-prob), D < 0
        v8f d = __builtin_amdgcn_wmma_f32_16x16x4_f32(
            /*neg_a=*/false, a, /*neg_b=*/false, bv,
            /*c_mod=*/(short)0, c0, /*reuse_a=*/false, /*reuse_b=*/false);

#pragma unroll
        for (int i = 0; i < 8; ++i)
            acc[i] += __builtin_amdgcn_exp2f(d[i]);   // exp(log_prob), safe: <= e^-0.92
    }

    // Row sums are striped across 16 lanes per half; reduce within each half.
#pragma unroll
    for (int m = 1; m < 16; m <<= 1) {
#pragma unroll
        for (int i = 0; i < 8; ++i)
            acc[i] += __shfl_xor(acc[i], m, 32);
    }

    // lanes 0-15 hold totals for rows 0..7 (slots 0..7); lanes 16-31 rows 8..15
    float llsum = 0.0f;
#pragma unroll
    for (int i = 0; i < 8; ++i)
        llsum += __builtin_amdgcn_logf(acc[i]) * 0.6931471805599453f;

    float partial = (lid == 0) ? llsum : 0.0f;     // keep lanes 0 and 16 only
    partial += __shfl_xor(partial, 16, 32);        // lane 0 now has all 16 rows
    if (lane == 0)
        atomicAdd(out, partial * (-1.0f / (float)(BATCH * NPTS)));
}

extern "C" void kernel_launch(void* const* d_in, const int* in_sizes, int n_in,
                              void* d_out, int out_size, void* d_ws, size_t ws_size,
                              hipStream_t stream) {
    (void)in_sizes; (void)n_in; (void)out_size; (void)d_ws; (void)ws_size;
    const float* pred = (const float*)d_in[0];   // (4, 4096, 3) f32
    const float* gt   = (const float*)d_in[1];   // (4, 4096, 3) f32
    float* out        = (float*)d_out;           // scalar f32 loss

    gmm_zero_out<<<1, 32, 0, stream>>>(out);
    gmm_loss_kernel<<<BATCH * BLOCKS_PER_BATCH, BLOCK_THREADS, 0, stream>>>(pred, gt, out);
}